// GaussianSetModule_33595234189980
// MI455X (gfx1250) — compile-verified
//
#include <hip/hip_runtime.h>
#include <stdint.h>

// Memory-bound streaming kernel (0.3 FLOP/byte): ~704 MB traffic => ~30us floor
// at 23.3 TB/s HBM. No GEMM structure => WMMA inapplicable; the CDNA5 feature
// that pays here is async global->LDS DMA to stage the 192B/point color rows,
// overlapped with the quaternion/covariance VALU math.

#define TILE 256
#define ROWF 52            // padded LDS row stride in floats (48 data + 4 pad)

typedef _Float16 f16;

__global__ __launch_bounds__(TILE)
void gaussian_pack_kernel(const float* __restrict__ positions,
                          const float* __restrict__ scales,
                          const float* __restrict__ rotations,
                          const float* __restrict__ opacity,
                          const float* __restrict__ colors,
                          f16* __restrict__ gout,   // N x 10 f16
                          f16* __restrict__ cout,   // N x 48 f16
                          int* __restrict__ npts,
                          int N)
{
    __shared__ float smem[TILE * ROWF];   // 53,248 B LDS tile

    const int tid = threadIdx.x;
    const int p0  = blockIdx.x * TILE;
    const int p   = p0 + tid;
    const int pl  = (p < N) ? p : (N - 1);   // clamped index for loads

    // ---------- Phase 1: kick off async DMA of this block's color tile ----------
    // colors tile = TILE*192B = 48KB, moved as 3072 x 16B chunks, 12 per thread.
    {
        const char* cbase = (const char*)colors;
        // low 32 bits of a flat shared-aperture address == wave-relative LDS byte offset
        unsigned sbase = (unsigned)(uintptr_t)&smem[0];
        const long long tile_byte = (long long)p0 * 192;
        const long long max_byte  = (long long)N * 192 - 16;
#pragma unroll
        for (int i = 0; i < 12; ++i) {
            int chunk = tid + i * TILE;          // 0..3071
            int pt    = chunk / 12;              // point within tile
            int part  = chunk - pt * 12;         // 16B part within 192B row
            long long gb = tile_byte + (long long)chunk * 16;
            if (gb > max_byte) gb = max_byte;    // tail-safe clamp
            unsigned loff = sbase + (unsigned)(pt * (ROWF * 4) + part * 16);
            uint64_t ga   = (uint64_t)(cbase + gb);
            asm volatile("global_load_async_to_lds_b128 %0, %1, off"
                         :: "v"(loff), "v"(ga)
                         : "memory");
        }
    }

    // ---------- Phase 2: per-point math, overlapped with the DMA ----------
    float4 q = ((const float4*)rotations)[pl];            // (w,x,y,z)
    float px = positions[3 * pl + 0];
    float py = positions[3 * pl + 1];
    float pz = positions[3 * pl + 2];
    float scx = scales[3 * pl + 0];
    float scy = scales[3 * pl + 1];
    float scz = scales[3 * pl + 2];
    float opv = opacity[pl];

    float d   = q.x * q.x + q.y * q.y + q.z * q.z + q.w * q.w;
    float inv = 1.0f / fmaxf(sqrtf(d), 1e-12f);
    float w = q.x * inv, x = q.y * inv, y = q.z * inv, z = q.w * inv;

    float ww = w * w, xx = x * x, yy = y * y, zz = z * z;
    float wx = w * x, wy = w * y, wz = w * z;
    float xy = x * y, xz = x * z, yz = y * z;

    float R00 = ww + xx - yy - zz, R01 = 2.f * (xy - wz), R02 = 2.f * (xz + wy);
    float R10 = 2.f * (xy + wz), R11 = ww - xx + yy - zz, R12 = 2.f * (yz - wx);
    float R20 = 2.f * (xz - wy), R21 = 2.f * (yz + wx), R22 = ww - xx - yy + zz;

    float s2x = __expf(2.f * scx);    // (exp(s))^2 == exp(2s)
    float s2y = __expf(2.f * scy);
    float s2z = __expf(2.f * scz);

    float C00 = R00 * R00 * s2x + R01 * R01 * s2y + R02 * R02 * s2z;
    float C01 = R00 * R10 * s2x + R01 * R11 * s2y + R02 * R12 * s2z;
    float C02 = R00 * R20 * s2x + R01 * R21 * s2y + R02 * R22 * s2z;
    float C11 = R10 * R10 * s2x + R11 * R11 * s2y + R12 * R12 * s2z;
    float C12 = R10 * R20 * s2x + R11 * R21 * s2y + R12 * R22 * s2z;
    float C22 = R20 * R20 * s2x + R21 * R21 * s2y + R22 * R22 * s2z;

    float sg = 1.f / (1.f + __expf(-opv));

    if (p < N) {
        union { f16 h[10]; unsigned u[5]; } g;
        g.h[0] = (f16)px;  g.h[1] = (f16)py;  g.h[2] = (f16)pz;  g.h[3] = (f16)sg;
        g.h[4] = (f16)C00; g.h[5] = (f16)C01; g.h[6] = (f16)C02;
        g.h[7] = (f16)C11; g.h[8] = (f16)C12; g.h[9] = (f16)C22;
        unsigned* dst = (unsigned*)(gout + (size_t)p * 10);  // 20B row, 4B aligned
#pragma unroll
        for (int i = 0; i < 5; ++i) dst[i] = g.u[i];
    }

    // ---------- Phase 3: consume LDS tile, emit permuted fp16 colors ----------
#if __has_builtin(__builtin_amdgcn_s_wait_asynccnt)
    __builtin_amdgcn_s_wait_asynccnt(0);
#else
    asm volatile("s_wait_asynccnt 0" ::: "memory");
#endif
    __syncthreads();

    if (p < N) {
        const float* row = &smem[tid * ROWF];   // stride 52 -> 2-way bank conflicts max
        union { f16 h[48]; uint4 v[6]; } o;
        o.h[0] = (f16)row[0];
        o.h[1] = (f16)row[1];
        o.h[2] = (f16)row[2];
#pragma unroll
        for (int k = 0; k < 15; ++k) {          // out[3+3k+c] = colors[3+15c+k]
            o.h[3 + 3 * k + 0] = (f16)row[3 + k];
            o.h[3 + 3 * k + 1] = (f16)row[18 + k];
            o.h[3 + 3 * k + 2] = (f16)row[33 + k];
        }
        uint4* dst = (uint4*)(cout + (size_t)p * 48);  // 96B row, 16B aligned
#pragma unroll
        for (int i = 0; i < 6; ++i) dst[i] = o.v[i];
    }

    if (p == 0) *npts = N;
}

extern "C" void kernel_launch(void* const* d_in, const int* in_sizes, int n_in,
                              void* d_out, int out_size, void* d_ws, size_t ws_size,
                              hipStream_t stream) {
    // setup_inputs order: camera, time, positions, scales, rotations, opacity, colors
    const float* positions = (const float*)d_in[2];
    const float* scales    = (const float*)d_in[3];
    const float* rotations = (const float*)d_in[4];
    const float* opacity   = (const float*)d_in[5];
    const float* colors    = (const float*)d_in[6];
    const int N = in_sizes[2] / 3;

    // Outputs flat in return order: gaussian_f16 (N*10 f16), out (N*48 f16),
    // num_points (1 x int32) appended.
    f16* gout = (f16*)d_out;
    f16* cout = gout + (size_t)N * 10;
    int* npts = (int*)(cout + (size_t)N * 48);

    const int blocks = (N + TILE - 1) / TILE;
    gaussian_pack_kernel<<<blocks, TILE, 0, stream>>>(
        positions, scales, rotations, opacity, colors, gout, cout, npts, N);
}